// SAGE_22505628631097
// MI455X (gfx1250) — compile-verified
//
#include <hip/hip_runtime.h>

// GraphSAGE 2-layer forward for MI455X (gfx1250, wave32).
// Dense parts use V_WMMA_F32_16X16X4_F32 (fp32 matrix pipe, reference precision).
// Mean-normalization is fused into GEMM A-staging; the final 128->2 classifier
// is fused into the layer-2 GEMM epilogue via LDS.

typedef float v2f __attribute__((ext_vector_type(2)));
typedef float v8f __attribute__((ext_vector_type(8)));

#define DFEAT 128  // NFEAT == NHID == 128

// ---------------- utility: zero a float region (grid-stride) ----------------
__global__ void zero_f32(float* __restrict__ p, long n) {
  long i = (long)blockIdx.x * blockDim.x + threadIdx.x;
  long stride = (long)gridDim.x * blockDim.x;
  for (; i < n; i += stride) p[i] = 0.0f;
}

// ---------------- pack a 128x128 weight matrix into WMMA fragment order ------
// For V_WMMA_F32_16X16X4_F32, lane L of a wave needs B-fragment
//   { W[k0][n], W[k0+1][n] }  with  n = ntile*16 + (L&15), k0 = ks*4 + (L>>4)*2.
// Packed layout: P[(ntile*32 + ks)*32 + lane] as float2  -> coalesced b64 loads.
__global__ void pack_w128(const float* __restrict__ W, float* __restrict__ P) {
  int idx = blockIdx.x * blockDim.x + threadIdx.x;   // 0 .. 8191
  if (idx >= 8 * 32 * 32) return;
  int lane  = idx & 31;
  int ks    = (idx >> 5) & 31;
  int ntile = idx >> 10;
  int k0 = ks * 4 + (lane >> 4) * 2;
  int n  = ntile * 16 + (lane & 15);
  float2 v;
  v.x = W[k0 * DFEAT + n];
  v.y = W[(k0 + 1) * DFEAT + n];
  reinterpret_cast<float2*>(P)[idx] = v;
}

// ---------------- edge scatter: acc[dst] += feat[src]; deg[dst] += 1 ---------
// 32 lanes per edge, each lane moves 4 floats (float4 gather, 4 f32 atomics).
// Accumulator is L2-resident (25.6 MB << 192 MB L2), so atomics stay on-chip.
__global__ void scatter_deg(const float* __restrict__ feat,
                            const int* __restrict__ src,
                            const int* __restrict__ dst,
                            float* __restrict__ acc,
                            float* __restrict__ deg, int E) {
  long gid = (long)blockIdx.x * blockDim.x + threadIdx.x;
  int e    = (int)(gid >> 5);
  int lane = (int)(gid & 31);
  if (e >= E) return;
  int s = src[e];
  int d = dst[e];
  float4 v = reinterpret_cast<const float4*>(feat + (long)s * DFEAT)[lane];
  float* o = acc + (long)d * DFEAT + lane * 4;
  atomicAdd(o + 0, v.x);
  atomicAdd(o + 1, v.y);
  atomicAdd(o + 2, v.z);
  atomicAdd(o + 3, v.w);
  if (lane == 0) atomicAdd(deg + d, 1.0f);
}

// ---------------- fused SAGE layer GEMM -------------------------------------
// 16x128 output tile per 256-thread block (8 waves; wave w owns N-tile w).
//   tile = As*Ws + (Asum/deg)*Wn + bias      (mean fused into LDS staging)
// MODE 1: apply relu, store tile to OutH      (layer 1)
// MODE 2: no relu; project tile through Wf (128->2) via LDS, store to OutC
//         (layer 2 + classifier fused; the hidden tile never touches HBM)
template <int MODE>
__global__ __launch_bounds__(256) void sage_gemm(
    const float* __restrict__ As, const float* __restrict__ Asum,
    const float* __restrict__ deg,
    const float* __restrict__ Ps, const float* __restrict__ Pn,
    const float* __restrict__ bias, float* __restrict__ OutH,
    const float* __restrict__ Wf, const float* __restrict__ bf,
    float* __restrict__ OutC) {
  __shared__ float ldsS[16 * DFEAT];
  __shared__ float ldsN[16 * DFEAT];
  const int  t       = threadIdx.x;
  const long rowbase = (long)blockIdx.x * 16;

  {  // stage 16x128 of each A operand: 512 float4 each, 2 per thread.
     // Neighbor rows are scaled by 1/max(deg,1) on the way in (fused mean).
    const float4* gs = reinterpret_cast<const float4*>(As + rowbase * DFEAT);
    const float4* gn = reinterpret_cast<const float4*>(Asum + rowbase * DFEAT);
    float4* ls = reinterpret_cast<float4*>(ldsS);
    float4* ln = reinterpret_cast<float4*>(ldsN);
    const int r1 = t >> 5;        // row of float4 index t      (0..7)
    const int r2 = 8 + (t >> 5);  // row of float4 index t+256  (8..15)
    float s1 = 1.0f / fmaxf(deg[rowbase + r1], 1.0f);
    float s2 = 1.0f / fmaxf(deg[rowbase + r2], 1.0f);
    float4 n1 = gn[t], n2 = gn[t + 256];
    ls[t]       = gs[t];
    ls[t + 256] = gs[t + 256];
    n1.x *= s1; n1.y *= s1; n1.z *= s1; n1.w *= s1;
    n2.x *= s2; n2.y *= s2; n2.z *= s2; n2.w *= s2;
    ln[t]       = n1;
    ln[t + 256] = n2;
  }
  __syncthreads();

  const int lane = t & 31;
  const int wid  = t >> 5;      // N-tile 0..7
  const int m    = lane & 15;   // A row within tile / output col within tile
  const int kh   = lane >> 4;   // K half-select per fragment layout
  const float2* ps = reinterpret_cast<const float2*>(Ps) + wid * 1024;
  const float2* pn = reinterpret_cast<const float2*>(Pn) + wid * 1024;

  v8f c = {0.f, 0.f, 0.f, 0.f, 0.f, 0.f, 0.f, 0.f};
#pragma unroll 8
  for (int ks = 0; ks < 32; ++ks) {
    const int k0 = ks * 4 + kh * 2;
    float2 as2 = *reinterpret_cast<const float2*>(&ldsS[m * DFEAT + k0]);
    float2 an2 = *reinterpret_cast<const float2*>(&ldsN[m * DFEAT + k0]);
    float2 bs2 = ps[ks * 32 + lane];
    float2 bn2 = pn[ks * 32 + lane];
    v2f a_s = {as2.x, as2.y};
    v2f a_n = {an2.x, an2.y};
    v2f b_s = {bs2.x, bs2.y};
    v2f b_n = {bn2.x, bn2.y};
    c = __builtin_amdgcn_wmma_f32_16x16x4_f32(false, a_s, false, b_s,
                                              (short)0, c, false, false);
    c = __builtin_amdgcn_wmma_f32_16x16x4_f32(false, a_n, false, b_n,
                                              (short)0, c, false, false);
  }

  // C/D layout: VGPR r -> row (r + 8*(lane>>4)), col = lane&15 within tile
  const int   col = wid * 16 + m;
  const float bv  = bias[col];
  const int   rb  = kh * 8;

  if (MODE == 1) {
#pragma unroll
    for (int r = 0; r < 8; ++r) {
      OutH[(rowbase + rb + r) * DFEAT + col] = fmaxf(c[r] + bv, 0.0f);
    }
  } else {
    // Classifier fusion: dump biased tile into (now free) ldsS, then 32
    // threads compute the 16x2 output tile. Row-skewed k index avoids the
    // 16-way bank conflict of stride-128 rows.
    __syncthreads();  // all waves done reading A tiles
#pragma unroll
    for (int r = 0; r < 8; ++r) {
      ldsS[(rb + r) * DFEAT + col] = c[r] + bv;
    }
    __syncthreads();
    if (t < 32) {
      const int row = t >> 1;
      const int cls = t & 1;
      float a = bf[cls];
      const int skew = row * 8;
#pragma unroll 8
      for (int kk = 0; kk < DFEAT; ++kk) {
        const int k = (kk + skew) & (DFEAT - 1);
        a += ldsS[row * DFEAT + k] * Wf[k * 2 + cls];
      }
      OutC[(rowbase + row) * 2 + cls] = a;
    }
  }
}

// ---------------------------------------------------------------------------
extern "C" void kernel_launch(void* const* d_in, const int* in_sizes, int n_in,
                              void* d_out, int out_size, void* d_ws,
                              size_t ws_size, hipStream_t stream) {
  const float* x    = (const float*)d_in[0];
  const int*   esrc = (const int*)d_in[1];
  const int*   edst = (const int*)d_in[2];
  const float* W1s  = (const float*)d_in[3];
  const float* W1n  = (const float*)d_in[4];
  const float* b1   = (const float*)d_in[5];
  const float* W2s  = (const float*)d_in[6];
  const float* W2n  = (const float*)d_in[7];
  const float* b2   = (const float*)d_in[8];
  const float* Wf   = (const float*)d_in[9];
  const float* bf   = (const float*)d_in[10];
  float* out = (float*)d_out;

  const int  N  = in_sizes[0] / DFEAT;   // 50000
  const int  E  = in_sizes[1];           // 800000
  const long NM = (long)N * DFEAT;

  float* ws  = (float*)d_ws;
  float* acc = ws;                 // NM   (neighbor sum; scaled on GEMM load)
  float* deg = acc + NM;           // N
  float* h1  = deg + N;            // NM
  float* P1s = h1 + NM;            // 16384 floats each (packed 128x128 W)
  float* P1n = P1s + 16384;
  float* P2s = P1n + 16384;
  float* P2n = P2s + 16384;

  // pack all 4 layer weights into WMMA fragment order (L2-resident, 64KB each)
  pack_w128<<<32, 256, 0, stream>>>(W1s, P1s);
  pack_w128<<<32, 256, 0, stream>>>(W1n, P1n);
  pack_w128<<<32, 256, 0, stream>>>(W2s, P2s);
  pack_w128<<<32, 256, 0, stream>>>(W2n, P2n);

  const int  rowTiles = N / 16;                       // 3125 (exact)
  const long sblocks  = ((long)E * 32 + 255) / 256;

  // ---- layer 1: h1 = relu(x@W1s + mean_neigh(x)@W1n + b1) ----
  zero_f32<<<2048, 256, 0, stream>>>(acc, NM + N);
  scatter_deg<<<(int)sblocks, 256, 0, stream>>>(x, esrc, edst, acc, deg, E);
  sage_gemm<1><<<rowTiles, 256, 0, stream>>>(x, acc, deg, P1s, P1n, b1, h1,
                                             nullptr, nullptr, nullptr);

  // ---- layer 2 + classifier: out = (h1@W2s + mean_neigh(h1)@W2n + b2)@Wf+bf
  zero_f32<<<2048, 256, 0, stream>>>(acc, NM + N);
  scatter_deg<<<(int)sblocks, 256, 0, stream>>>(h1, esrc, edst, acc, deg, E);
  sage_gemm<2><<<rowTiles, 256, 0, stream>>>(h1, acc, deg, P2s, P2n, b2,
                                             nullptr, Wf, bf, out);
}